// DeformableAttention1D_34780645163255
// MI455X (gfx1250) — compile-verified
//
#include <hip/hip_runtime.h>
#include <hip/hip_bf16.h>
#include <math.h>
#include <stdint.h>

typedef __attribute__((ext_vector_type(2))) float v2f;
typedef __attribute__((ext_vector_type(8))) float v8f;

// ---------------------------------------------------------------------------
// Generic fp32 WMMA GEMM:  C[M,N] = A[M,K] @ B[K,N] (+ bias[M])
// One wave per 16x16 C tile, grid.z = group (pointer offsets).
// A 16x4 f32 layout: lanes 0-15 row=lane, vgpr{0,1}=K{0,1};
//                    lanes 16-31 row=lane-16, vgpr{0,1}=K{2,3}.
// B 4x16 f32 layout: vgpr v, half h -> K row = v + 2*h, col = lane&15.
// C/D: vgpr j -> M = j + 8*(lane>>4), N = lane&15.
// ---------------------------------------------------------------------------
__global__ void gemm16_f32_wmma(const float* __restrict__ A,
                                const float* __restrict__ B,
                                const float* __restrict__ bias,
                                float* __restrict__ C,
                                int M, int N, int K,
                                long aStride, long bStride, long cStride) {
    const int g = blockIdx.z;
    A += (long)g * aStride;
    B += (long)g * bStride;
    C += (long)g * cStride;
    const int mBase = blockIdx.y * 16;
    const int nBase = blockIdx.x * 16;
    const int lane  = threadIdx.x;
    const int half  = lane >> 4;
    const int l16   = lane & 15;

    v8f acc = {};
    const int row = mBase + l16;
    const int col = nBase + l16;
    for (int k0 = 0; k0 < K; k0 += 4) {
        const int ka = k0 + 2 * half;
        v2f a, b;
        a.x = A[(long)row * K + ka];
        a.y = A[(long)row * K + ka + 1];
        b.x = B[(long)ka * N + col];
        b.y = B[(long)(ka + 1) * N + col];
        acc = __builtin_amdgcn_wmma_f32_16x16x4_f32(false, a, false, b,
                                                    (short)0, acc, false, false);
    }
#pragma unroll
    for (int j = 0; j < 8; ++j) {
        const int m = mBase + j + 8 * half;
        float v = acc[j];
        if (bias) v += bias[m];
        C[(long)m * N + nBase + l16] = v;
    }
}

// ---------------------------------------------------------------------------
// Offset path: depthwise conv (k=6, stride=4, pad=1) + bias + erf-GELU,
// channel projection, tanh*4, then vgrid_scaled.  One block per (bg, j).
// ---------------------------------------------------------------------------
__global__ void offsets_kernel(const float* __restrict__ q,     // (256,1024)
                               const float* __restrict__ wdw,   // (64,6)
                               const float* __restrict__ bdw,   // (64)
                               const float* __restrict__ wproj, // (64)
                               float* __restrict__ vgs) {       // (4,256)
    const int bg = blockIdx.y;
    const int j  = blockIdx.x;
    const int c  = threadIdx.x;                 // 0..63
    const float* row = q + (long)(bg * 64 + c) * 1024;
    const int base = 4 * j - 1;
    float s = 0.f;
#pragma unroll
    for (int t = 0; t < 6; ++t) {
        const int p = base + t;
        const float xv = (p >= 0 && p < 1024) ? row[p] : 0.f;
        s += wdw[c * 6 + t] * xv;
    }
    s += bdw[c];
    const float ge = 0.5f * s * (1.f + erff(s * 0.70710678118654752440f));
    __shared__ float red[64];
    red[c] = ge * wproj[c];
    __syncthreads();
    for (int off = 32; off > 0; off >>= 1) {
        if (c < off) red[c] += red[c + off];
        __syncthreads();
    }
    if (c == 0) {
        const float offv = tanhf(red[0]) * 4.0f;
        const float vg = (float)j + offv;
        vgs[bg * 256 + j] = 2.0f * vg / 255.0f - 1.0f;
    }
}

// ---------------------------------------------------------------------------
// 1-D bilinear grid sample (zeros padding, align_corners=False).
// ---------------------------------------------------------------------------
__global__ void gridsample_kernel(const float* __restrict__ x,   // (256,1024)
                                  const float* __restrict__ vgs, // (4,256)
                                  float* __restrict__ kv) {      // (256,256)
    const int bg = blockIdx.y;
    const int j  = blockIdx.x;
    const int c  = threadIdx.x;                 // 0..63
    const float g = vgs[bg * 256 + j];
    const float p = (g + 1.0f) * 512.0f - 0.5f;
    const float i0f = floorf(p);
    const float w1 = p - i0f, w0 = 1.f - w1;
    const int i0 = (int)i0f, i1 = i0 + 1;
    const float* row = x + (long)(bg * 64 + c) * 1024;
    const float f0 = (i0 >= 0 && i0 < 1024) ? row[i0] : 0.f;
    const float f1 = (i1 >= 0 && i1 < 1024) ? row[i1] : 0.f;
    kv[(long)(bg * 64 + c) * 256 + j] = f0 * w0 + f1 * w1;
}

// ---------------------------------------------------------------------------
// CPB relative-position MLP (dominant ~9 GFLOP), register-blocked.
// GEMM orientation: out^T[o, pair] = W2[o,i] @ h1^T[i, pair].
//  - A = W2, held in registers (4 M-tiles x 16 k-steps = 128 VGPRs),
//    reused across a 16-q loop per wave (16x operand reuse).
//  - B = layer-1 output, generated directly in WMMA B layout by VALU
//    (pos is per-lane: lane&15 = j within tile; w1/b1 loads are
//    q-invariant so LICM hoists them).
//  - C: M = output channel -> layer-3 is a per-lane dot over VGPRs plus
//    ONE shfl_xor(16) across halves; stores coalesced over j.
// One wave per (bg, jt, q-chunk of 16).
// ---------------------------------------------------------------------------
__global__ void cpb_kernel(const float* __restrict__ vgs, // (4,256)
                           const float* __restrict__ w1,  // (64)
                           const float* __restrict__ b1,  // (64)
                           const float* __restrict__ w2,  // (64,64) [o][i]
                           const float* __restrict__ w3,  // (2,64)
                           const float* __restrict__ b2,  // (64)
                           const float* __restrict__ b3,  // (2)
                           float* __restrict__ biasOut) { // (8,1024,256)
    const int jt   = blockIdx.x;   // 0..15
    const int qc   = blockIdx.y;   // 0..63 (16 q each)
    const int bg   = blockIdx.z;   // 0..3
    const int lane = threadIdx.x;
    const int half = lane >> 4;
    const int l16  = lane & 15;

    // A = W2 in WMMA A layout, register resident.
    v2f aw[4][16];
#pragma unroll
    for (int t = 0; t < 4; ++t) {
        const int o = t * 16 + l16;
#pragma unroll
        for (int s = 0; s < 16; ++s) {
            const int ka = 4 * s + 2 * half;
            aw[t][s].x = w2[o * 64 + ka];
            aw[t][s].y = w2[o * 64 + ka + 1];
        }
    }
    const float vg  = vgs[bg * 256 + jt * 16 + l16];  // per-lane j coordinate
    const float b30 = b3[0], b31 = b3[1];

    for (int qi = 0; qi < 16; ++qi) {
        const int qpos = qc * 16 + qi;
        const float gq  = 2.0f * (float)qpos / 1023.0f - 1.0f;
        const float d   = gq - vg;
        const float pos = copysignf(log1pf(fabsf(d)), d);

        v8f acc[4] = {};
#pragma unroll
        for (int s = 0; s < 16; ++s) {
            const int ka = 4 * s + 2 * half;
            v2f b;                                    // layer-1 + relu, B layout
            b.x = fmaxf(fmaf(pos, w1[ka],     b1[ka]),     0.f);
            b.y = fmaxf(fmaf(pos, w1[ka + 1], b1[ka + 1]), 0.f);
#pragma unroll
            for (int t = 0; t < 4; ++t)
                acc[t] = __builtin_amdgcn_wmma_f32_16x16x4_f32(
                    false, aw[t][s], false, b, (short)0, acc[t], false, false);
        }
        // layer-2 bias+relu, layer-3 dot over 32 output channels per lane
        float p0 = 0.f, p1 = 0.f;
#pragma unroll
        for (int t = 0; t < 4; ++t)
#pragma unroll
            for (int vj = 0; vj < 8; ++vj) {
                const int o = t * 16 + 8 * half + vj;
                const float h2 = fmaxf(acc[t][vj] + b2[o], 0.f);
                p0 = fmaf(h2, w3[o], p0);
                p1 = fmaf(h2, w3[64 + o], p1);
            }
        p0 += __shfl_xor(p0, 16, 32);               // combine the two halves
        p1 += __shfl_xor(p1, 16, 32);
        if (half == 0) {
            const int j = jt * 16 + l16;            // coalesced over lanes
            biasOut[((long)(bg * 2 + 0) * 1024 + qpos) * 256 + j] = p0 + b30;
            biasOut[((long)(bg * 2 + 1) * 1024 + qpos) * 256 + j] = p1 + b31;
        }
    }
}

// ---------------------------------------------------------------------------
// Fused attention: per (head, 16-q tile) block of 4 waves.
//  - V head slab (32x256 f32 = 32KB) is async-copied to LDS with
//    global_load_async_to_lds_b128 at kernel entry; the copy overlaps the
//    entire sim-WMMA + softmax phase and is fenced with s_wait_asynccnt.
//  - sim (16x256, K=32) via WMMA + CPB bias -> LDS -> row softmax ->
//    out (16x32, K=256) via WMMA from LDS (K split across waves) -> reduce.
//  - LDS rows padded (260 / 258) so per-lane B/A fetches hit distinct banks.
// ---------------------------------------------------------------------------
__global__ void attn_kernel(const float* __restrict__ Q,   // (256,1024) unscaled
                            const float* __restrict__ Kf,  // (256,256)
                            const float* __restrict__ Vf,  // (256,256)
                            const float* __restrict__ Bb,  // (8,1024,256)
                            float* __restrict__ Oh) {      // (256,1024)
    const int qt  = blockIdx.x;   // 0..63
    const int h   = blockIdx.y;   // 0..7
    const int tid = threadIdx.x;  // 0..127
    const int w    = tid >> 5;
    const int lane = tid & 31;
    const int half = lane >> 4;
    const int l16  = lane & 15;
    const float scale = 0.17677669529663687f;    // 32^-0.5

    __shared__ float s_attn[16][260];            // padded: bank-conflict free
    __shared__ float s_v[32][258];               // padded V slab
    __shared__ float s_part[4][16][32];

    // ---- kick off async V copy: thread -> (row, quarter), 16 x b128 each ----
    {
        const int r   = tid >> 2;                // 0..31 (d channel)
        const int qtr = (tid & 3) * 64;          // dword offset in row
        const float* gp = Vf + (long)(h * 32 + r) * 256 + qtr;
        const uint32_t lbase = (uint32_t)(uintptr_t)&s_v[r][qtr];
#pragma unroll
        for (int i = 0; i < 16; ++i) {
            asm volatile("global_load_async_to_lds_b128 %0, %1, off"
                         :: "v"(lbase + i * 16), "v"(gp + i * 4)
                         : "memory");
        }
    }

    // ---- sim tile: wave w owns j-tiles w*4 .. w*4+3 (overlaps async copy) ----
    v8f acc[4] = {};
    const int qcol = qt * 16 + l16;
#pragma unroll
    for (int s = 0; s < 8; ++s) {               // K = 32
        const int ka = 4 * s + 2 * half;
        v2f a;
        a.x = Q[(long)(h * 32 + ka) * 1024 + qcol] * scale;
        a.y = Q[(long)(h * 32 + ka + 1) * 1024 + qcol] * scale;
#pragma unroll
        for (int t = 0; t < 4; ++t) {
            const int jcol = (w * 4 + t) * 16 + l16;
            v2f b;
            b.x = Kf[(long)(h * 32 + ka) * 256 + jcol];
            b.y = Kf[(long)(h * 32 + ka + 1) * 256 + jcol];
            acc[t] = __builtin_amdgcn_wmma_f32_16x16x4_f32(false, a, false, b,
                                                           (short)0, acc[t],
                                                           false, false);
        }
    }
#pragma unroll
    for (int t = 0; t < 4; ++t) {
        const int jcol = (w * 4 + t) * 16 + l16;
#pragma unroll
        for (int vj = 0; vj < 8; ++vj) {
            const int m = vj + 8 * half;
            const int qpos = qt * 16 + m;
            s_attn[m][jcol] = acc[t][vj] + Bb[((long)h * 1024 + qpos) * 256 + jcol];
        }
    }
    __syncthreads();

    // ---- stable softmax over j (one thread per q row) ----
    if (tid < 16) {
        float mx = -3.402823466e38f;
        for (int j = 0; j < 256; ++j) mx = fmaxf(mx, s_attn[tid][j]);
        float sum = 0.f;
        for (int j = 0; j < 256; ++j) {
            const float e = expf(s_attn[tid][j] - mx);
            s_attn[tid][j] = e;
            sum += e;
        }
        const float inv = 1.0f / sum;
        for (int j = 0; j < 256; ++j) s_attn[tid][j] *= inv;
    }
    // fence this wave's async V copy, then barrier => V fully in LDS
    asm volatile("s_wait_asynccnt 0x0" ::: "memory");
    __syncthreads();

    // ---- out = attn (16x256) @ V^T (256x32) from LDS; wave w: K slice ----
    v8f oacc[2] = {};
#pragma unroll
    for (int s = 0; s < 16; ++s) {
        const int ka = (w * 16 + s) * 4 + 2 * half;
        v2f a;
        a.x = s_attn[l16][ka];
        a.y = s_attn[l16][ka + 1];
#pragma unroll
        for (int t = 0; t < 2; ++t) {
            const int dcol = t * 16 + l16;
            v2f b;                                // B[k=j][n=d] = V[d][j]
            b.x = s_v[dcol][ka];
            b.y = s_v[dcol][ka + 1];
            oacc[t] = __builtin_amdgcn_wmma_f32_16x16x4_f32(false, a, false, b,
                                                            (short)0, oacc[t],
                                                            false, false);
        }
    }
#pragma unroll
    for (int t = 0; t < 2; ++t)
#pragma unroll
        for (int vj = 0; vj < 8; ++vj)
            s_part[w][vj + 8 * half][t * 16 + l16] = oacc[t][vj];
    __syncthreads();

    for (int idx = tid; idx < 512; idx += 128) {
        const int m = idx >> 5, n = idx & 31;
        const float sum = s_part[0][m][n] + s_part[1][m][n] +
                          s_part[2][m][n] + s_part[3][m][n];
        Oh[(long)(h * 32 + n) * 1024 + qt * 16 + m] = sum;
    }
}

// ---------------------------------------------------------------------------
extern "C" void kernel_launch(void* const* d_in, const int* in_sizes, int n_in,
                              void* d_out, int out_size, void* d_ws, size_t ws_size,
                              hipStream_t stream) {
    const float* x      = (const float*)d_in[0];   // (256,1024)
    const float* wq     = (const float*)d_in[1];   // (4,64,64)
    const float* wk     = (const float*)d_in[2];
    const float* wv     = (const float*)d_in[3];
    const float* wo     = (const float*)d_in[4];   // (256,256)
    const float* bo     = (const float*)d_in[5];   // (256)
    const float* wdw    = (const float*)d_in[6];   // (64,1,6)
    const float* bdw    = (const float*)d_in[7];   // (64)
    const float* wproj  = (const float*)d_in[8];   // (64)
    const float* cpbw1  = (const float*)d_in[9];   // (64,1)
    const float* cpbb1  = (const float*)d_in[10];  // (64)
    const float* cpbw2  = (const float*)d_in[11];  // (64,64)
    const float* cpbb2  = (const float*)d_in[12];  // (64)
    const float* cpbw3  = (const float*)d_in[13];  // (2,64)
    const float* cpbb3  = (const float*)d_in[14];  // (2)
    float* out = (float*)d_out;                    // (256,1024)

    // workspace carve-up (floats): ~11.3 MB total
    float* ws     = (float*)d_ws;
    float* q      = ws;                  // 256*1024
    float* kv     = q      + 262144;     // 256*256
    float* kbuf   = kv     + 65536;      // 256*256
    float* vbuf   = kbuf   + 65536;      // 256*256
    float* vgs    = vbuf   + 65536;      // 4*256
    float* biasb  = vgs    + 1024;       // 8*1024*256
    float* oheads = biasb  + 2097152;    // 256*1024

    // 1) q = grouped conv1x1 (4 GEMMs 64x1024x64)
    gemm16_f32_wmma<<<dim3(64, 4, 4), 32, 0, stream>>>(
        wq, x, nullptr, q, 64, 1024, 64, 4096, 65536, 65536);

    // 2) offsets -> vgrid_scaled
    offsets_kernel<<<dim3(256, 4), 64, 0, stream>>>(q, wdw, bdw, wproj, vgs);

    // 3) deformable bilinear gather -> kv
    gridsample_kernel<<<dim3(256, 4), 64, 0, stream>>>(x, vgs, kv);

    // 4) k, v = grouped conv1x1 on kv (4 GEMMs 64x256x64)
    gemm16_f32_wmma<<<dim3(16, 4, 4), 32, 0, stream>>>(
        wk, kv, nullptr, kbuf, 64, 256, 64, 4096, 16384, 16384);
    gemm16_f32_wmma<<<dim3(16, 4, 4), 32, 0, stream>>>(
        wv, kv, nullptr, vbuf, 64, 256, 64, 4096, 16384, 16384);

    // 5) CPB bias MLP: register-blocked W2, 16-q loop per wave
    cpb_kernel<<<dim3(16, 64, 4), 32, 0, stream>>>(
        vgs, cpbw1, cpbb1, cpbw2, cpbw3, cpbb2, cpbb3, biasb);

    // 6) fused sim + bias + softmax + attn@v (async V staging to LDS)
    attn_kernel<<<dim3(64, 8), 128, 0, stream>>>(q, kbuf, vbuf, biasb, oheads);

    // 7) output projection (+ bias): out = wo @ oheads + bo
    gemm16_f32_wmma<<<dim3(64, 16, 1), 32, 0, stream>>>(
        wo, oheads, bo, out, 256, 1024, 256, 0, 0, 0);
}